// SRPooler_40218073760348
// MI455X (gfx1250) — compile-verified
//
#include <hip/hip_runtime.h>
#include <stdint.h>

// ROI-Align for MI455X (gfx1250).
// Bandwidth-bound gather (<1 flop/byte): TDM stages per-roi 3D tiles
// (x-contig, row stride W, channel stride H*W) into LDS, double-buffered on
// TENSORcnt. Separable bilinear: y-contraction on VALU -> T in LDS, then
// x-contraction as a real GEMM via V_WMMA_F32_16X16X4_F32 (M=16 channels,
// N=16 (7 used) bin columns, K=ix). 1/(S*S) and validity folded into Wx.

#define PP 7
#define SS 2
#define RSCALE 0.125f
#define IH 100
#define IW 100
#define NC 256
#define NR 512
#define CH 16
#define NCHUNK (NC / CH)
#define MAXK 26                 // bin span <= 6.5*25/7 + 2 < 26
#define KXPAD 28                // pad Kx to 7 WMMA K-steps of 4
#define XTILE (CH * MAXK * MAXK)

typedef __attribute__((ext_vector_type(4))) unsigned int v4u;
typedef __attribute__((ext_vector_type(4))) int v4i;
typedef __attribute__((ext_vector_type(8))) int v8i;
typedef __attribute__((ext_vector_type(2))) float v2f;
typedef __attribute__((ext_vector_type(8))) float v8f;

#if defined(__HIP_DEVICE_COMPILE__) && __has_builtin(__builtin_amdgcn_tensor_load_to_lds) && __has_builtin(__builtin_amdgcn_s_wait_tensorcnt)
#define USE_TDM 1
#else
#define USE_TDM 0
#endif

#if defined(__HIP_DEVICE_COMPILE__) && __has_builtin(__builtin_amdgcn_wmma_f32_16x16x4_f32)
#define USE_WMMA 1
#else
#define USE_WMMA 0
#endif

#if __has_include(<hip/amd_detail/amd_gfx1250_TDM.h>)
#define TDM_SIX_ARGS 1          // amdgpu-toolchain clang-23: 6-arg builtin
#else
#define TDM_SIX_ARGS 0          // ROCm 7.2 clang-22: 5-arg builtin
#endif

__device__ __forceinline__ int imin(int a, int b) { return a < b ? a : b; }
__device__ __forceinline__ int imax(int a, int b) { return a > b ? a : b; }

#if USE_TDM
__device__ __forceinline__ unsigned rfl(unsigned v) {
  return (unsigned)__builtin_amdgcn_readfirstlane((int)v);
}

// 3D tile: dim0 = Kx (contig), dim1 = Ky (stride W), dim2 = CH (stride H*W).
__device__ __forceinline__ void tdm_issue(const float* xg, int bIdx, int c0,
                                          int y0, int x0, int Ky, int Kx,
                                          unsigned ldsOff) {
  uint64_t ga = (uint64_t)(uintptr_t)xg +
                4ull * ((uint64_t)(unsigned)(bIdx * NC + c0) * (IH * IW) +
                        (unsigned)(y0 * IW + x0));
  v4u g0;
  g0[0] = rfl(1u);                                          // count=1, user mode
  g0[1] = rfl(ldsOff);                                      // lds_addr (bytes)
  g0[2] = rfl((unsigned)ga);                                // global_addr[31:0]
  g0[3] = rfl(((unsigned)(ga >> 32) & 0x1FFFFFFu) | (2u << 30)); // addr[56:32] | type=2
  v8i g1;
  g1[0] = (int)rfl(0x00020000u);                            // data_size=4B, mask=0
  g1[1] = (int)rfl((unsigned)Kx << 16);                     // tensor_dim0[15:0]
  g1[2] = (int)rfl((unsigned)Ky << 16);                     // dim0 hi=0 | tensor_dim1 lo
  g1[3] = (int)rfl((unsigned)Kx << 16);                     // dim1 hi=0 | tile_dim0
  g1[4] = (int)rfl((unsigned)Ky | ((unsigned)CH << 16));    // tile_dim1 | tile_dim2
  g1[5] = (int)rfl((unsigned)IW);                           // tensor_dim0_stride lo
  g1[6] = (int)rfl(((unsigned)(IH * IW) & 0xFFFFu) << 16);  // d0s hi=0 | dim1_stride lo
  g1[7] = (int)rfl((unsigned)(IH * IW) >> 16);              // tensor_dim1_stride hi
  v4i g2; g2[0] = (int)rfl((unsigned)CH); g2[1] = 0; g2[2] = 0; g2[3] = 0; // tensor_dim2
  v4i g3; g3[0] = 0; g3[1] = 0; g3[2] = 0; g3[3] = 0;
#if TDM_SIX_ARGS
  v8i g4; g4[0]=0; g4[1]=0; g4[2]=0; g4[3]=0; g4[4]=0; g4[5]=0; g4[6]=0; g4[7]=0;
  __builtin_amdgcn_tensor_load_to_lds(g0, g1, g2, g3, g4, 0);
#else
  __builtin_amdgcn_tensor_load_to_lds(g0, g1, g2, g3, 0);
#endif
}
#endif

__global__ __launch_bounds__(256) void roialign_tdm_wmma_kernel(
    const float* __restrict__ xg, const float* __restrict__ rois,
    float* __restrict__ out) {
  extern __shared__ unsigned char smem_raw[];
  float* Xbuf = (float*)smem_raw;                 // 2 * XTILE floats
  float* T    = Xbuf + 2 * XTILE;                 // CH*PP*KXPAD floats
  float* Wx   = T + CH * PP * KXPAD;              // 16*KXPAD floats (rows>=7 stay 0)
  int*   ilo  = (int*)(Wx + 16 * KXPAD);          // ylo[14],yhi[14],xlo[14],xhi[14]
  float* fwt  = (float*)(ilo + 56);               // ywlo[14],ywhi[14],xwlo[14],xwhi[14]

  const int tid  = threadIdx.x;
  const int wave = tid >> 5;
  const int r    = blockIdx.x;

  const int   bIdx = (int)rois[r * 5 + 0];
  const float sxv  = rois[r * 5 + 1] * RSCALE - 0.5f;
  const float syv  = rois[r * 5 + 2] * RSCALE - 0.5f;
  const float binw = (rois[r * 5 + 3] * RSCALE - 0.5f - sxv) / (float)PP;
  const float binh = (rois[r * 5 + 4] * RSCALE - 0.5f - syv) / (float)PP;

  for (int i = tid; i < 16 * KXPAD; i += 256) Wx[i] = 0.0f;

  // Per-axis sample terms (matches reference _axis_terms).
  if (tid < 2 * PP * SS) {
    const bool  isX   = tid >= PP * SS;
    const int   j     = isX ? tid - PP * SS : tid;
    const float start = isX ? sxv : syv;
    const float bsz   = isX ? binw : binh;
    const float Lf    = isX ? (float)IW : (float)IH;
    const int   py = j / SS, s = j % SS;
    const float cg = start + ((float)py + ((float)s + 0.5f) / (float)SS) * bsz;
    const bool  valid = (cg > -1.0f) && (cg < Lf);
    const float cc = fminf(fmaxf(cg, 0.0f), Lf - 1.0f);
    int lo = imin(imax((int)floorf(cc), 0), (int)Lf - 1);
    int hi = imin(lo + 1, (int)Lf - 1);
    const float frac = cc - (float)lo;
    const int base = isX ? 28 : 0;
    ilo[base + j]      = lo;
    ilo[base + 14 + j] = hi;
    fwt[base + j]      = valid ? 1.0f - frac : 0.0f;
    fwt[base + 14 + j] = valid ? frac : 0.0f;
  }
  __syncthreads();

  // Bounding box (uniform across the block).
  int y0 = IH - 1, y1 = 0, x0 = IW - 1, x1 = 0;
  for (int j = 0; j < PP * SS; ++j) {
    y0 = imin(y0, ilo[j]);      y1 = imax(y1, ilo[14 + j]);
    x0 = imin(x0, ilo[28 + j]); x1 = imax(x1, ilo[42 + j]);
  }
  const int Ky = imin(imax(y1 - y0 + 1, 1), MAXK);
  const int Kx = imin(imax(x1 - x0 + 1, 1), MAXK);

  // Build Wx[px][ix] (scatter of x-weights, 1/(S*S) folded in); 1 thread/row.
  if (tid < PP) {
    for (int s = 0; s < SS; ++s) {
      const int j   = tid * SS + s;
      const int cl  = imin(imax(ilo[28 + j] - x0, 0), KXPAD - 1);
      const int chh = imin(imax(ilo[42 + j] - x0, 0), KXPAD - 1);
      Wx[tid * KXPAD + cl]  += 0.25f * fwt[28 + j];
      Wx[tid * KXPAD + chh] += 0.25f * fwt[42 + j];
    }
  }

#if USE_TDM
  if (wave == 0)
    tdm_issue(xg, bIdx, 0, y0, x0, Ky, Kx, (unsigned)(uintptr_t)Xbuf);
#endif

  for (int chnk = 0; chnk < NCHUNK; ++chnk) {
    const int bi = chnk & 1;
    float* Xb = Xbuf + bi * XTILE;
#if USE_TDM
    if (wave == 0) {
      if (chnk + 1 < NCHUNK) {
        tdm_issue(xg, bIdx, (chnk + 1) * CH, y0, x0, Ky, Kx,
                  (unsigned)(uintptr_t)(Xbuf + ((chnk + 1) & 1) * XTILE));
        __builtin_amdgcn_s_wait_tensorcnt(1);   // chunk `chnk` arrived (in-order)
      } else {
        __builtin_amdgcn_s_wait_tensorcnt(0);
      }
    }
    __syncthreads();
#else
    {
      const float* src = xg + (size_t)(unsigned)(bIdx * NC + chnk * CH) * (IH * IW);
      const int tot = CH * Ky * Kx;
      for (int i = tid; i < tot; i += 256) {
        const int c = i / (Ky * Kx), rm = i % (Ky * Kx);
        Xb[i] = src[(size_t)c * (IH * IW) + (size_t)(y0 + rm / Kx) * IW + (x0 + rm % Kx)];
      }
    }
    __syncthreads();
#endif

    // Stage A: y-contraction. T[c][py][ix] = sum_s wy_lo*X[yl] + wy_hi*X[yh]
    for (int i = tid; i < CH * PP * KXPAD; i += 256) {
      const int c  = i / (PP * KXPAD);
      const int rm = i % (PP * KXPAD);
      const int py = rm / KXPAD;
      const int ix = rm % KXPAD;
      float acc = 0.0f;
      if (ix < Kx) {
        const float* Xc = Xb + c * (Ky * Kx) + ix;
#pragma unroll
        for (int s = 0; s < SS; ++s) {
          const int j  = py * SS + s;
          const int il = imin(imax(ilo[j] - y0, 0), Ky - 1);
          const int ih = imin(imax(ilo[14 + j] - y0, 0), Ky - 1);
          acc += fwt[j] * Xc[il * Kx] + fwt[14 + j] * Xc[ih * Kx];
        }
      }
      T[i] = acc;   // zero in the pad; Wx pad is zero too
    }
    __syncthreads();

    // Stage B: x-contraction as GEMM. D[c][px] = sum_ix T[c][py][ix]*Wx[px][ix]
#if USE_WMMA
    if (wave < PP) {                       // wave w handles py = w, full EXEC
      const int lane = tid & 31;
      const int m    = lane & 15;          // A row (channel) == B col (px)
      const int koff = (lane < 16) ? 0 : 2;// A/B VGPR K-split per ISA layout
      const int py   = wave;
      const float* Trow = T  + (m * PP + py) * KXPAD;
      const float* Wrow = Wx + m * KXPAD;
      v8f acc = {0.f, 0.f, 0.f, 0.f, 0.f, 0.f, 0.f, 0.f};
#pragma unroll
      for (int kb = 0; kb < KXPAD / 4; ++kb) {
        const int ix = kb * 4 + koff;
        v2f av; av[0] = Trow[ix]; av[1] = Trow[ix + 1];   // A 16x4 f32
        v2f bv; bv[0] = Wrow[ix]; bv[1] = Wrow[ix + 1];   // B 4x16 f32
        acc = __builtin_amdgcn_wmma_f32_16x16x4_f32(false, av, false, bv,
                                                    (short)0, acc, false, false);
      }
      const int px = m;
      if (px < PP) {
        const int cbase = chnk * CH + ((lane < 16) ? 0 : 8);
#pragma unroll
        for (int i = 0; i < 8; ++i)   // D vgpr i: M = i (+8 for upper lanes)
          out[(((size_t)r * NC + (cbase + i)) * PP + py) * PP + px] = acc[i];
      }
    }
#else
    for (int i = tid; i < CH * PP * PP; i += 256) {
      const int c = i / (PP * PP), rm = i % (PP * PP);
      const int py = rm / PP, px = rm % PP;
      float s = 0.0f;
      for (int ix = 0; ix < KXPAD; ++ix)
        s += T[(c * PP + py) * KXPAD + ix] * Wx[px * KXPAD + ix];
      out[(((size_t)r * NC + (chnk * CH + c)) * PP + py) * PP + px] = s;
    }
#endif
    __syncthreads();   // T / Xbuf reuse fence
  }
}

extern "C" void kernel_launch(void* const* d_in, const int* in_sizes, int n_in,
                              void* d_out, int out_size, void* d_ws, size_t ws_size,
                              hipStream_t stream) {
  (void)in_sizes; (void)n_in; (void)out_size; (void)d_ws; (void)ws_size;
  const float* x    = (const float*)d_in[0];
  const float* rois = (const float*)d_in[1];
  float* out = (float*)d_out;
  const size_t smem = (size_t)(2 * XTILE + CH * PP * KXPAD + 16 * KXPAD) * sizeof(float)
                    + 56 * sizeof(int) + 56 * sizeof(float);
  roialign_tdm_wmma_kernel<<<dim3(NR), dim3(256), smem, stream>>>(x, rois, out);
}